// JointLoss_11897059410010
// MI455X (gfx1250) — compile-verified
//
#include <hip/hip_runtime.h>
#include <hip/hip_bf16.h>

typedef float v2f __attribute__((ext_vector_type(2)));
typedef float v4f __attribute__((ext_vector_type(4)));
typedef float v8f __attribute__((ext_vector_type(8)));

#define B_N 8192
#define C_N 8192
#define D_N 256

#define MARGIN_F 0.3f
#define SIMM_F 0.85f

// GEMM tiling
#define BM 128   // block rows (8 waves x 16)
#define BN 64    // block cols (4 x 16 per wave)
#define BK 32    // K chunk
#define LDST 36  // padded LDS row stride (floats): conflict-free + 16B aligned rows

// async global->LDS path (gfx1250): bypass VGPRs, track with ASYNCcnt
#if defined(__has_builtin)
#if __has_builtin(__builtin_amdgcn_global_load_async_to_lds_b128) && \
    __has_builtin(__builtin_amdgcn_s_wait_asynccnt)
#define USE_ASYNC_LDS 1
#endif
#endif
#ifndef USE_ASYNC_LDS
#define USE_ASYNC_LDS 0
#endif

#if USE_ASYNC_LDS
// builtin expects pointer-to-int4 (b128 payload) in global (AS1) / LDS (AS3)
typedef int vint4_t __attribute__((vector_size(16)));
typedef vint4_t __attribute__((address_space(1))) * as1_v4i_ptr;
typedef vint4_t __attribute__((address_space(3))) * as3_v4i_ptr;
#endif

// workspace layout (float indices)
#define OFF_POS 0
#define OFF_A2  16
#define OFF_F2S (OFF_A2 + C_N)
#define OFF_F2T (OFF_F2S + B_N)
#define OFF_SS  (OFF_F2T + B_N)
#define OFF_CS  (OFF_SS + B_N)
#define OFF_ST  (OFF_CS + B_N)
#define OFF_CT  (OFF_ST + B_N)
#define WS_FLOATS (OFF_CT + B_N)

__global__ void init_ws_kernel(float* __restrict__ ws) {
    int i = blockIdx.x * blockDim.x + threadIdx.x;
    if (i < WS_FLOATS) ws[i] = 0.0f;
}

// one wave per row: out[row] = sum_k x[row,k]^2
__global__ void rownorm_kernel(const float* __restrict__ x, float* __restrict__ out, int rows) {
    int wave = (blockIdx.x * blockDim.x + threadIdx.x) >> 5;
    int lane = threadIdx.x & 31;
    if (wave >= rows) return;
    const float* r = x + (size_t)wave * D_N;
    float s = 0.0f;
    for (int i = lane * 4; i < D_N; i += 128) {
        v4f v = *(const v4f*)(r + i);
        s += v.x * v.x + v.y * v.y + v.z * v.z + v.w * v.w;
    }
    for (int m = 16; m >= 1; m >>= 1) s += __shfl_xor(s, m, 32);
    if (lane == 0) out[wave] = s;
}

// one wave per row: atomic sum of ||f_b - a_{label_b}||^2
__global__ void posloss_kernel(const float* __restrict__ F, const float* __restrict__ A,
                               const int* __restrict__ LAB, float* __restrict__ sum_pos) {
    int wave = (blockIdx.x * blockDim.x + threadIdx.x) >> 5;
    int lane = threadIdx.x & 31;
    if (wave >= B_N) return;
    const float* f = F + (size_t)wave * D_N;
    const float* a = A + (size_t)LAB[wave] * D_N;
    float s = 0.0f;
    for (int i = lane * 4; i < D_N; i += 128) {
        v4f vf = *(const v4f*)(f + i);
        v4f va = *(const v4f*)(a + i);
        float dx = vf.x - va.x, dy = vf.y - va.y, dz = vf.z - va.z, dw = vf.w - va.w;
        s += dx * dx + dy * dy + dz * dz + dw * dw;
    }
    for (int m = 16; m >= 1; m >>= 1) s += __shfl_xor(s, m, 32);
    if (lane == 0) atomicAdd(sum_pos, s);
}

// stage one BMxBK tile of F and BNxBK tile of A into LDS buffers
__device__ __forceinline__ void stage_tiles(const float* __restrict__ F,
                                            const float* __restrict__ A,
                                            float* Asb, float* Bsb,
                                            int rowBase, int colBase, int kb, int tid) {
#if USE_ASYNC_LDS
    #pragma unroll
    for (int it = 0; it < 4; ++it) {
        int p = tid + it * 256;
        int r = p >> 3;
        int cv = p & 7;
        const float* g = F + (size_t)(rowBase + r) * D_N + kb + cv * 4;
        __builtin_amdgcn_global_load_async_to_lds_b128(
            (as1_v4i_ptr)g, (as3_v4i_ptr)&Asb[r * LDST + cv * 4], 0, 0);
    }
    #pragma unroll
    for (int it = 0; it < 2; ++it) {
        int p = tid + it * 256;
        int r = p >> 3;
        int cv = p & 7;
        const float* g = A + (size_t)(colBase + r) * D_N + kb + cv * 4;
        __builtin_amdgcn_global_load_async_to_lds_b128(
            (as1_v4i_ptr)g, (as3_v4i_ptr)&Bsb[r * LDST + cv * 4], 0, 0);
    }
#else
    #pragma unroll
    for (int it = 0; it < 4; ++it) {
        int p = tid + it * 256;
        int r = p >> 3;
        int cv = p & 7;
        v4f v = *(const v4f*)(F + (size_t)(rowBase + r) * D_N + kb + cv * 4);
        *(v4f*)(&Asb[r * LDST + cv * 4]) = v;
    }
    #pragma unroll
    for (int it = 0; it < 2; ++it) {
        int p = tid + it * 256;
        int r = p >> 3;
        int cv = p & 7;
        v4f v = *(const v4f*)(A + (size_t)(colBase + r) * D_N + kb + cv * 4);
        *(v4f*)(&Bsb[r * LDST + cv * 4]) = v;
    }
#endif
}

__device__ __forceinline__ void async_wait_all() {
#if USE_ASYNC_LDS
    __builtin_amdgcn_s_wait_asynccnt(0);
#endif
}

// F:[rows x D] x A:[C x D]^T GEMM via WMMA f32 16x16x4, double-buffered LDS,
// masked-hinge epilogue. rs[m] += hinge sum, rc[m] += masked count.
__global__ __launch_bounds__(256)
void hinge_gemm_kernel(const float* __restrict__ F, const float* __restrict__ A,
                       const float* __restrict__ SIM, const int* __restrict__ LAB,
                       const float* __restrict__ f2, const float* __restrict__ a2,
                       float* __restrict__ rs, float* __restrict__ rc, int use_labels) {
    __shared__ float As[2][BM * LDST];
    __shared__ float Bs[2][BN * LDST];

    const int tid = threadIdx.x;
    const int wave = tid >> 5;   // 0..7
    const int lane = tid & 31;
    const int hi = lane >> 4;    // 0/1 half
    const int lo = lane & 15;

    const int rowBase = blockIdx.y * BM;
    const int colBase = blockIdx.x * BN;

    v8f acc[4] = {};

    // prologue: stage first chunk
    stage_tiles(F, A, As[0], Bs[0], rowBase, colBase, 0, tid);
    async_wait_all();
    __syncthreads();

    int cur = 0;
    for (int kb = 0; kb < D_N; kb += BK) {
        int nxt = cur ^ 1;
        // prefetch next chunk into the other buffer while computing this one
        if (kb + BK < D_N)
            stage_tiles(F, A, As[nxt], Bs[nxt], rowBase, colBase, kb + BK, tid);

        const float* ap = &As[cur][(wave * 16 + lo) * LDST];
        const float* bp = &Bs[cur][lo * LDST];
        #pragma unroll
        for (int kk = 0; kk < BK; kk += 4) {
            // A frag per ISA layout: lanes 0-15 carry K=kk,kk+1 ; lanes 16-31 carry K=kk+2,kk+3
            v2f a = *(const v2f*)(ap + kk + 2 * hi);
            #pragma unroll
            for (int t = 0; t < 4; ++t) {
                v2f b = *(const v2f*)(bp + (t * 16) * LDST + kk + 2 * hi);
                acc[t] = __builtin_amdgcn_wmma_f32_16x16x4_f32(
                    /*neg_a=*/false, a, /*neg_b=*/false, b,
                    /*c_mod=*/(short)0, acc[t], /*reuse_a=*/false, /*reuse_b=*/false);
            }
        }

        // next buffer complete + everyone done reading current before swap
        async_wait_all();
        __syncthreads();
        cur = nxt;
    }

    // epilogue: d = f2[m] + a2[n] - 2*g ; hinge under mask; per-row reduce
    const int mBaseW = rowBase + wave * 16;
    float fr[8];
    #pragma unroll
    for (int r = 0; r < 8; ++r) fr[r] = f2[mBaseW + r + 8 * hi];

    float rsum[8], rcnt[8];
    #pragma unroll
    for (int r = 0; r < 8; ++r) { rsum[r] = 0.0f; rcnt[r] = 0.0f; }

    #pragma unroll
    for (int t = 0; t < 4; ++t) {
        int n = colBase + t * 16 + lo;
        float a2n = a2[n];
        #pragma unroll
        for (int r = 0; r < 8; ++r) {
            int m = mBaseW + r + 8 * hi;
            float dsq = fr[r] + a2n - 2.0f * acc[t][r];
            float h = fmaxf(0.0f, MARGIN_F - dsq);
            float simv = SIM[(size_t)m * C_N + n];
            bool msk = simv > SIMM_F;
            if (use_labels) msk = msk && (LAB[m] != n);
            if (msk) { rsum[r] += h; rcnt[r] += 1.0f; }
        }
    }

    // reduce over the 16 N-lanes of each half (rows differ between halves)
    #pragma unroll
    for (int r = 0; r < 8; ++r) {
        float s = rsum[r], c = rcnt[r];
        for (int m2 = 8; m2 >= 1; m2 >>= 1) {
            s += __shfl_xor(s, m2, 32);
            c += __shfl_xor(c, m2, 32);
        }
        rsum[r] = s; rcnt[r] = c;
    }
    if (lo == 0) {
        #pragma unroll
        for (int r = 0; r < 8; ++r) {
            if (rcnt[r] > 0.0f) {
                int m = mBaseW + r + 8 * hi;
                atomicAdd(&rs[m], rsum[r]);
                atomicAdd(&rc[m], rcnt[r]);
            }
        }
    }
}

__global__ void finalize_kernel(const float* __restrict__ ws, float* __restrict__ out) {
    __shared__ float st[256];
    __shared__ float stt[256];
    float tot = 0.0f, terms = 0.0f;
    for (int b = threadIdx.x; b < B_N; b += 256) {
        float c = ws[OFF_CS + b];
        if (c > 0.0f) { tot += ws[OFF_SS + b] / c; terms += 1.0f; }
        float ct = ws[OFF_CT + b];
        if (ct > 0.0f) { tot += ws[OFF_ST + b] / ct; terms += 1.0f; }
    }
    st[threadIdx.x] = tot;
    stt[threadIdx.x] = terms;
    __syncthreads();
    for (int s = 128; s > 0; s >>= 1) {
        if ((int)threadIdx.x < s) {
            st[threadIdx.x] += st[threadIdx.x + s];
            stt[threadIdx.x] += stt[threadIdx.x + s];
        }
        __syncthreads();
    }
    if (threadIdx.x == 0) {
        float total = st[0] + ws[OFF_POS];
        out[0] = total / ((float)B_N + stt[0]);
    }
}

extern "C" void kernel_launch(void* const* d_in, const int* in_sizes, int n_in,
                              void* d_out, int out_size, void* d_ws, size_t ws_size,
                              hipStream_t stream) {
    const float* F    = (const float*)d_in[0];  // features [B,D]
    const float* A    = (const float*)d_in[1];  // agents [C,D]
    const int*   LAB  = (const int*)d_in[2];    // labels [B]
    const float* SIM  = (const float*)d_in[3];  // similarity [B,C]
    const float* FT   = (const float*)d_in[4];  // features_target [B,D]
    const float* SIMT = (const float*)d_in[5];  // similarity_target [B,C]
    float* ws  = (float*)d_ws;
    float* out = (float*)d_out;

    init_ws_kernel<<<(WS_FLOATS + 255) / 256, 256, 0, stream>>>(ws);
    rownorm_kernel<<<C_N / 8, 256, 0, stream>>>(A, ws + OFF_A2, C_N);
    rownorm_kernel<<<B_N / 8, 256, 0, stream>>>(F, ws + OFF_F2S, B_N);
    rownorm_kernel<<<B_N / 8, 256, 0, stream>>>(FT, ws + OFF_F2T, B_N);
    posloss_kernel<<<B_N / 8, 256, 0, stream>>>(F, A, LAB, ws + OFF_POS);

    dim3 grid(C_N / BN, B_N / BM);
    hinge_gemm_kernel<<<grid, 256, 0, stream>>>(F, A, SIM, LAB,
                                                ws + OFF_F2S, ws + OFF_A2,
                                                ws + OFF_SS, ws + OFF_CS, 1);
    hinge_gemm_kernel<<<grid, 256, 0, stream>>>(FT, A, SIMT, LAB,
                                                ws + OFF_F2T, ws + OFF_A2,
                                                ws + OFF_ST, ws + OFF_CT, 0);
    finalize_kernel<<<1, 256, 0, stream>>>(ws, out);
}